// MuCoMiD_31860067402122
// MI455X (gfx1250) — compile-verified
//
#include <hip/hip_runtime.h>
#include <hip/hip_bf16.h>
#include <math.h>

#define EMB 256
#define HID 128
#define N_M 2048
#define N_D 1024
#define N_P 50000
#define E_M 65536
#define E_D 32768
#define E_P 1600000
#define P_MP 500000
#define P_DP 500000
#define P_LBL 250000

typedef __attribute__((ext_vector_type(2))) float v2f;
typedef __attribute__((ext_vector_type(8))) float v8f;

static inline unsigned ceil_div(unsigned a, unsigned b) { return (a + b - 1) / b; }

// ---------------------------------------------------------------------------
// deg[i] = 1.0 (self-loop weight)
// ---------------------------------------------------------------------------
__global__ void deg_init_kernel(float* __restrict__ deg, int n) {
    int i = blockIdx.x * blockDim.x + threadIdx.x;
    if (i < n) deg[i] = 1.0f;
}

// deg[col] += w  over all edges (hardware f32 atomic -> global_atomic_add_f32)
__global__ void deg_accum_kernel(const int* __restrict__ el, const float* __restrict__ ew,
                                 float* __restrict__ deg, int E) {
    int e = blockIdx.x * blockDim.x + threadIdx.x;
    if (e < E) unsafeAtomicAdd(&deg[el[2 * e + 1]], ew[e]);
}

// in place: deg -> dis = deg>0 ? rsqrt(max(deg,1e-12)) : 0
__global__ void deg_rsqrt_kernel(float* __restrict__ deg, int n) {
    int i = blockIdx.x * blockDim.x + threadIdx.x;
    if (i < n) {
        float d = deg[i];
        deg[i] = (d > 0.0f) ? rsqrtf(fmaxf(d, 1e-12f)) : 0.0f;
    }
}

// ---------------------------------------------------------------------------
// XW = X[M,256] @ W[256,128] in fp32 via V_WMMA_F32_16X16X4_F32.
// Block = 128 threads = 4 waves; wave w handles M-tile (blockIdx.x*4+w),
// accumulating all 8 N-tiles (16x128 output) over K=256 in steps of 4.
// A layout (16x4 f32): lanes 0-15 M rows, VGPR0/1 = K0/K1; lanes 16-31 K2/K3.
// B layout (4x16 f32): row (=K) striped across lanes within a VGPR.
// C/D layout: VGPR v, lanes 0-15 -> M=v, lanes 16-31 -> M=v+8, N=lane&15.
// M is a multiple of 16 for all three graphs, so EXEC is all-1s per live wave.
// ---------------------------------------------------------------------------
__global__ __launch_bounds__(128) void gemm_xw_wmma_kernel(
        const float* __restrict__ X, const float* __restrict__ W,
        float* __restrict__ XW, int M) {
    const int wave = threadIdx.x >> 5;          // uniform within wave
    const int lane = threadIdx.x & 31;
    const int mtile = blockIdx.x * 4 + wave;
    if (mtile * 16 >= M) return;                // whole-wave uniform exit

    const int l15 = lane & 15;
    const int hi  = lane >> 4;                  // 0: lanes 0-15, 1: lanes 16-31
    const float* xrow = X + (size_t)(mtile * 16 + l15) * EMB;

    v8f acc[8];
#pragma unroll
    for (int t = 0; t < 8; ++t) acc[t] = (v8f){0.f, 0.f, 0.f, 0.f, 0.f, 0.f, 0.f, 0.f};

    for (int k = 0; k < EMB; k += 4) {
        const int ka = k + hi * 2;
        v2f a;
        a.x = xrow[ka];
        a.y = xrow[ka + 1];
#pragma unroll
        for (int t = 0; t < 8; ++t) {
            v2f b;
            b.x = W[(size_t)ka * HID + t * 16 + l15];
            b.y = W[(size_t)(ka + 1) * HID + t * 16 + l15];
            acc[t] = __builtin_amdgcn_wmma_f32_16x16x4_f32(
                false, a, false, b, (short)0, acc[t], false, false);
        }
    }

#pragma unroll
    for (int t = 0; t < 8; ++t) {
#pragma unroll
        for (int v = 0; v < 8; ++v) {
            const int row = mtile * 16 + v + hi * 8;
            const int col = t * 16 + l15;
            XW[(size_t)row * HID + col] = acc[t][v];
        }
    }
}

// ---------------------------------------------------------------------------
// acc[i,:] = dis[i]^2 * xw[i,:]   (self-loop contribution; also inits acc)
// ---------------------------------------------------------------------------
__global__ void self_loop_init_kernel(const float* __restrict__ dis,
                                      const float* __restrict__ xw,
                                      float* __restrict__ acc, int n) {
    int i = blockIdx.x * blockDim.x + threadIdx.x;   // over n = N*HID
    if (i < n) {
        float d = dis[i >> 7];
        acc[i] = d * d * xw[i];
    }
}

// One wave per edge: acc[col,:] += dis[row]*w*dis[col] * xw[row,:]
// Each lane handles 4 consecutive floats (float4 load, 4 hw f32 atomics).
__global__ void edge_scatter_kernel(const int* __restrict__ el, const float* __restrict__ ew,
                                    const float* __restrict__ dis, const float* __restrict__ xw,
                                    float* __restrict__ acc, int E) {
    int gwave = (blockIdx.x * blockDim.x + threadIdx.x) >> 5;
    int lane  = threadIdx.x & 31;
    if (gwave >= E) return;
    int r = el[2 * gwave];
    int c = el[2 * gwave + 1];
    float norm = dis[r] * ew[gwave] * dis[c];
    float4 v = ((const float4*)(xw + (size_t)r * HID))[lane];
    float* dst = acc + (size_t)c * HID + lane * 4;
    unsafeAtomicAdd(dst + 0, norm * v.x);
    unsafeAtomicAdd(dst + 1, norm * v.y);
    unsafeAtomicAdd(dst + 2, norm * v.z);
    unsafeAtomicAdd(dst + 3, norm * v.w);
}

// in place: acc = relu(acc + b[col])
__global__ void bias_relu_kernel(float* __restrict__ acc, const float* __restrict__ b, int n) {
    int i = blockIdx.x * blockDim.x + threadIdx.x;
    if (i < n) acc[i] = fmaxf(acc[i] + b[i & (HID - 1)], 0.0f);
}

// ---------------------------------------------------------------------------
// One wave per pair: out[p] = sigmoid( dot(A[i,:]*B[j,:], W) + bias )
// lane handles 4 floats; cross-lane reduce via shfl_xor (wave32).
// ---------------------------------------------------------------------------
__global__ void pair_score_kernel(const int* __restrict__ pairs,
                                  const float* __restrict__ A, const float* __restrict__ B,
                                  const float* __restrict__ W, const float* __restrict__ bias,
                                  float* __restrict__ out, int P) {
    int p    = (blockIdx.x * blockDim.x + threadIdx.x) >> 5;
    int lane = threadIdx.x & 31;
    if (p >= P) return;
    int i = pairs[2 * p];
    int j = pairs[2 * p + 1];
    float4 a = ((const float4*)(A + (size_t)i * HID))[lane];
    float4 b = ((const float4*)(B + (size_t)j * HID))[lane];
    float4 w = ((const float4*)W)[lane];
    float s = a.x * b.x * w.x + a.y * b.y * w.y + a.z * b.z * w.z + a.w * b.w * w.w;
#pragma unroll
    for (int off = 16; off > 0; off >>= 1) s += __shfl_xor(s, off, 32);
    if (lane == 0) {
        float z = s + bias[0];
        out[p] = 1.0f / (1.0f + __expf(-z));
    }
}

// ---------------------------------------------------------------------------
// Host-side per-graph GCN pipeline
// ---------------------------------------------------------------------------
static void run_gcn(const float* x, const float* W, const float* b,
                    const int* el, const float* ew, int E, int N,
                    float* xw, float* acc, float* deg, hipStream_t stream) {
    const int BT = 256;
    deg_init_kernel<<<ceil_div(N, BT), BT, 0, stream>>>(deg, N);
    deg_accum_kernel<<<ceil_div(E, BT), BT, 0, stream>>>(el, ew, deg, E);
    deg_rsqrt_kernel<<<ceil_div(N, BT), BT, 0, stream>>>(deg, N);
    gemm_xw_wmma_kernel<<<ceil_div(N, 64), 128, 0, stream>>>(x, W, xw, N);
    self_loop_init_kernel<<<ceil_div(N * HID, BT), BT, 0, stream>>>(deg, xw, acc, N * HID);
    edge_scatter_kernel<<<ceil_div(E * 32, BT), BT, 0, stream>>>(el, ew, deg, xw, acc, E);
    bias_relu_kernel<<<ceil_div(N * HID, BT), BT, 0, stream>>>(acc, b, N * HID);
}

extern "C" void kernel_launch(void* const* d_in, const int* in_sizes, int n_in,
                              void* d_out, int out_size, void* d_ws, size_t ws_size,
                              hipStream_t stream) {
    const float* mirna_emb   = (const float*)d_in[0];
    const float* disease_emb = (const float*)d_in[1];
    const float* pcg_emb     = (const float*)d_in[2];
    const int*   m_el        = (const int*)d_in[3];
    const float* m_ew        = (const float*)d_in[4];
    const int*   d_el        = (const int*)d_in[5];
    const float* d_ew        = (const float*)d_in[6];
    const int*   p_el        = (const int*)d_in[7];
    const float* p_ew        = (const float*)d_in[8];
    const int*   mp_pairs    = (const int*)d_in[9];
    const int*   dp_pairs    = (const int*)d_in[10];
    const int*   lbl         = (const int*)d_in[11];
    const float* Wm          = (const float*)d_in[12];
    const float* Wd          = (const float*)d_in[13];
    const float* Wp          = (const float*)d_in[14];
    const float* bm          = (const float*)d_in[15];
    const float* bd          = (const float*)d_in[16];
    const float* bp          = (const float*)d_in[17];
    const float* W_assoc     = (const float*)d_in[18];
    const float* b_assoc     = (const float*)d_in[19];
    const float* W_mp        = (const float*)d_in[20];
    const float* b_mp        = (const float*)d_in[21];
    const float* W_dp        = (const float*)d_in[22];
    const float* b_dp        = (const float*)d_in[23];

    // workspace layout (floats); total ~13.7M floats (~55 MB)
    float* ws    = (float*)d_ws;
    float* xw_m  = ws; ws += (size_t)N_M * HID;
    float* xw_d  = ws; ws += (size_t)N_D * HID;
    float* xw_p  = ws; ws += (size_t)N_P * HID;
    float* acc_m = ws; ws += (size_t)N_M * HID;
    float* acc_d = ws; ws += (size_t)N_D * HID;
    float* acc_p = ws; ws += (size_t)N_P * HID;
    float* deg_m = ws; ws += N_M;
    float* deg_d = ws; ws += N_D;
    float* deg_p = ws; ws += N_P;

    run_gcn(mirna_emb,   Wm, bm, m_el, m_ew, E_M, N_M, xw_m, acc_m, deg_m, stream);
    run_gcn(disease_emb, Wd, bd, d_el, d_ew, E_D, N_D, xw_d, acc_d, deg_d, stream);
    run_gcn(pcg_emb,     Wp, bp, p_el, p_ew, E_P, N_P, xw_p, acc_p, deg_p, stream);

    float* out_assoc = (float*)d_out;
    float* out_mp    = out_assoc + P_LBL;
    float* out_dp    = out_mp + P_MP;

    const int BT = 256;
    pair_score_kernel<<<ceil_div(P_LBL * 32, BT), BT, 0, stream>>>(
        lbl, acc_m, acc_d, W_assoc, b_assoc, out_assoc, P_LBL);
    pair_score_kernel<<<ceil_div(P_MP * 32, BT), BT, 0, stream>>>(
        mp_pairs, acc_m, acc_p, W_mp, b_mp, out_mp, P_MP);
    pair_score_kernel<<<ceil_div(P_DP * 32, BT), BT, 0, stream>>>(
        dp_pairs, acc_d, acc_p, W_dp, b_dp, out_dp, P_DP);
}